// RelGNN_88648124990074
// MI455X (gfx1250) — compile-verified
//
#include <hip/hip_runtime.h>
#include <hip/hip_bf16.h>

typedef __attribute__((ext_vector_type(2))) float v2f;
typedef __attribute__((ext_vector_type(8))) float v8f;

#define EMB 32
#define HID 64

// ---------------------------------------------------------------------------
// x0[n,32] = shape_emb[sid[n]] + col_emb[cid[n]] + pos_emb[pid[n]]
// ---------------------------------------------------------------------------
__global__ void embed_kernel(const int* __restrict__ sid, const int* __restrict__ cid,
                             const int* __restrict__ pid,
                             const float* __restrict__ se, const float* __restrict__ ce,
                             const float* __restrict__ pe,
                             float* __restrict__ x0, int Nn) {
    long idx = (long)blockIdx.x * blockDim.x + threadIdx.x;
    if (idx >= (long)Nn * EMB) return;
    int n = (int)(idx >> 5);
    int j = (int)(idx & 31);
    x0[idx] = se[sid[n] * EMB + j] + ce[cid[n] * EMB + j] + pe[pid[n] * EMB + j];
}

// ---------------------------------------------------------------------------
// cnt[r*N + dst] += 1 for each edge (per-relation in-degree)
// ---------------------------------------------------------------------------
__global__ void count_kernel(const int* __restrict__ dst, const int* __restrict__ et,
                             float* __restrict__ cnt, int E, int Nn) {
    long e = (long)blockIdx.x * blockDim.x + threadIdx.x;
    if (e >= E) return;
    atomicAdd(&cnt[(size_t)et[e] * Nn + dst[e]], 1.0f);
}

// cnt -> 1/max(cnt,1) in place (hoists rcp out of the 6M-item scatter pass)
__global__ void inv_kernel(float* __restrict__ cnt, long n) {
    long idx = (long)blockIdx.x * blockDim.x + threadIdx.x;
    if (idx < n) cnt[idx] = 1.0f / fmaxf(cnt[idx], 1.0f);
}

// ---------------------------------------------------------------------------
// Fused RGCN transform GEMM:  X[N,K] @ [root | W0 | W1 | W2]  (K x 256)
//   chunk 0 (root) -> H[row, 0:64] + bias     (initializes H)
//   chunk r+1      -> Z[r, row, 0:64]
// Weight panel staged in LDS with k-pairs interleaved: element (k,c) at
// lds[(k>>1)*512 + c*2 + (k&1)], so each B fragment (k, k+1 at column c)
// is one contiguous 8B ds_load_b64 landing in an adjacent VGPR pair — no
// regrouping movs before the WMMA. One wave owns a 16x64 tile via 4x v8f
// accs, stepping K by 4 with V_WMMA_F32_16X16X4_F32. A-row index is clamped
// (not branched): the K-loop carries no EXEC manipulation; OOB lanes produce
// garbage accs whose stores are guarded.
// ---------------------------------------------------------------------------
__global__ void rgcn_gemm_kernel(const float* __restrict__ X,
                                 const float* __restrict__ root,
                                 const float* __restrict__ Wrel,
                                 const float* __restrict__ bias,
                                 float* __restrict__ H,
                                 float* __restrict__ Z,
                                 int Nn, int K) {
    extern __shared__ float lds[];               // [K/2][256][2]
    const int tid = threadIdx.x;

    // cooperative fill of the concatenated weight panel (interleaved pairs)
    for (int idx = tid; idx < K * 256; idx += blockDim.x) {
        int k = idx >> 8;
        int c = idx & 255;
        float v;
        if (c < 64) v = root[k * 64 + c];
        else        v = Wrel[(((c >> 6) - 1) * K + k) * 64 + (c & 63)];
        lds[(k >> 1) * 512 + c * 2 + (k & 1)] = v;
    }
    __syncthreads();

    const int lane    = tid & 31;
    const int wave    = tid >> 5;
    const int rowTile = wave >> 2;               // 0..1
    const int chunk   = wave & 3;                // 0..3 : root / rel0 / rel1 / rel2
    const int row0    = blockIdx.x * 32 + rowTile * 16;
    const int halfSel = lane >> 4;               // 0: K=0,1  1: K=2,3
    const int l15     = lane & 15;

    // clamped A row pointer: loop-invariant, no branches in the K-loop
    int rowA = row0 + l15;
    if (rowA > Nn - 1) rowA = Nn - 1;
    const float* __restrict__ Arow = X + (size_t)rowA * K;

    v8f acc[4];
#pragma unroll
    for (int n = 0; n < 4; ++n)
#pragma unroll
        for (int v = 0; v < 8; ++v) acc[n][v] = 0.0f;

    for (int k0 = 0; k0 < K; k0 += 4) {
        const int kb = k0 + (halfSel << 1);      // even
        v2f a;
        a.x = Arow[kb];
        a.y = Arow[kb + 1];
        const float* __restrict__ bbase = lds + (kb >> 1) * 512;
#pragma unroll
        for (int n = 0; n < 4; ++n) {
            const int col = chunk * 64 + n * 16 + l15;
            const v2f b = *(const v2f*)(bbase + col * 2);   // ds_load_b64
            acc[n] = __builtin_amdgcn_wmma_f32_16x16x4_f32(
                false, a, false, b, (short)0, acc[n], false, false);
        }
    }

    // store D: lane<16 -> M = v, lane>=16 -> M = v+8 ; N = lane&15
#pragma unroll
    for (int n = 0; n < 4; ++n) {
#pragma unroll
        for (int v = 0; v < 8; ++v) {
            const int M   = v + (halfSel << 3);
            const int row = row0 + M;
            if (row >= Nn) continue;
            const int c64 = n * 16 + l15;
            if (chunk == 0)
                H[(size_t)row * 64 + c64] = acc[n][v] + bias[c64];
            else
                Z[((size_t)(chunk - 1) * Nn + row) * 64 + c64] = acc[n][v];
        }
    }
}

// ---------------------------------------------------------------------------
// Edge scatter: h[dst] += z[et][src] * inv[et][dst]
// Grid-stride loop, 4 lanes per edge, 16 floats each, float4 gathers +
// f32 global atomics. Each lane prefetches its NEXT iteration's 64B gather
// slice (global_prefetch_b8) to hide random-access latency behind the
// current iteration's atomics.
// ---------------------------------------------------------------------------
__global__ void scatter_kernel(const int* __restrict__ src, const int* __restrict__ dst,
                               const int* __restrict__ et,
                               const float* __restrict__ Z, const float* __restrict__ inv,
                               float* __restrict__ H, int E, int Nn) {
    const long total  = (long)E * 4;
    const long stride = (long)gridDim.x * blockDim.x;
    for (long idx = (long)blockIdx.x * blockDim.x + threadIdx.x; idx < total; idx += stride) {
        // prefetch next iteration's gather row slice
        const long nidx = idx + stride;
        if (nidx < total) {
            const int ne = (int)(nidx >> 2);
            const int ns = src[ne];
            const int nr = et[ne];
            __builtin_prefetch(Z + (((size_t)nr * Nn + ns) * 64 + (nidx & 3) * 16), 0, 0);
        }

        const int e    = (int)(idx >> 2);
        const int part = (int)(idx & 3);
        const int s = src[e], d = dst[e], r = et[e];
        const float scale = inv[(size_t)r * Nn + d];
        const float4* zin = (const float4*)(Z + (((size_t)r * Nn + s) * 64 + part * 16));
        float* hout       = H + ((size_t)d * 64 + part * 16);
#pragma unroll
        for (int q = 0; q < 4; ++q) {
            float4 v = zin[q];
            atomicAdd(hout + q * 4 + 0, v.x * scale);
            atomicAdd(hout + q * 4 + 1, v.y * scale);
            atomicAdd(hout + q * 4 + 2, v.z * scale);
            atomicAdd(hout + q * 4 + 3, v.w * scale);
        }
    }
}

__global__ void relu_kernel(float* __restrict__ h, long n) {
    long idx = (long)blockIdx.x * blockDim.x + threadIdx.x;
    if (idx < n) h[idx] = fmaxf(h[idx], 0.0f);
}

// ---------------------------------------------------------------------------
// Global mean pool (atomic accumulate), then final [G,64]@[64,2] linear.
// ---------------------------------------------------------------------------
__global__ void pool_kernel(const float* __restrict__ h, const int* __restrict__ batch,
                            float* __restrict__ psum, float* __restrict__ pcnt, int Nn) {
    long idx = (long)blockIdx.x * blockDim.x + threadIdx.x;
    if (idx >= (long)Nn * 64) return;
    int n = (int)(idx >> 6);
    int j = (int)(idx & 63);
    int b = batch[n];
    atomicAdd(&psum[(size_t)b * 64 + j], h[idx]);
    if (j == 0) atomicAdd(&pcnt[b], 1.0f);
}

__global__ void final_kernel(const float* __restrict__ psum, const float* __restrict__ pcnt,
                             const float* __restrict__ lw, const float* __restrict__ lb,
                             float* __restrict__ out, int G) {
    int g = blockIdx.x * blockDim.x + threadIdx.x;
    if (g >= G) return;
    float inv = 1.0f / fmaxf(pcnt[g], 1.0f);
    float o0 = lb[0], o1 = lb[1];
#pragma unroll
    for (int j = 0; j < 64; ++j) {
        float v = psum[(size_t)g * 64 + j] * inv;
        o0 += v * lw[j * 2 + 0];
        o1 += v * lw[j * 2 + 1];
    }
    out[g * 2 + 0] = o0;
    out[g * 2 + 1] = o1;
}

// ---------------------------------------------------------------------------
extern "C" void kernel_launch(void* const* d_in, const int* in_sizes, int n_in,
                              void* d_out, int out_size, void* d_ws, size_t ws_size,
                              hipStream_t stream) {
    const int*   sid   = (const int*)  d_in[0];
    const int*   cid   = (const int*)  d_in[1];
    const int*   pid   = (const int*)  d_in[2];
    const int*   eidx  = (const int*)  d_in[3];   // [2,E]
    const int*   etyp  = (const int*)  d_in[4];
    const int*   batch = (const int*)  d_in[5];
    const float* se    = (const float*)d_in[7];
    const float* ce    = (const float*)d_in[8];
    const float* pe    = (const float*)d_in[9];
    const float* W1    = (const float*)d_in[10];
    const float* root1 = (const float*)d_in[11];
    const float* b1    = (const float*)d_in[12];
    const float* W2    = (const float*)d_in[13];
    const float* root2 = (const float*)d_in[14];
    const float* b2    = (const float*)d_in[15];
    const float* lw    = (const float*)d_in[16];
    const float* lb    = (const float*)d_in[17];
    float* out = (float*)d_out;

    const int N = in_sizes[0];
    const int E = in_sizes[4];
    const int G = out_size / 2;
    const int* esrc = eidx;
    const int* edst = eidx + E;

    // workspace layout (floats)
    float* ws   = (float*)d_ws;
    float* x0   = ws;                         // N*32
    float* z    = x0   + (size_t)N * 32;      // 3*N*64
    float* h1   = z    + (size_t)N * 192;     // N*64
    float* h2   = h1   + (size_t)N * 64;      // N*64
    float* cnt  = h2   + (size_t)N * 64;      // 3*N   (becomes inverse counts)
    float* psum = cnt  + (size_t)N * 3;       // G*64
    float* pcnt = psum + (size_t)G * 64;      // G

    hipMemsetAsync(cnt,  0, (size_t)N * 3 * sizeof(float), stream);
    hipMemsetAsync(psum, 0, (size_t)G * 64 * sizeof(float), stream);
    hipMemsetAsync(pcnt, 0, (size_t)G * sizeof(float), stream);

    const int TB = 256;
    // 1. node features
    embed_kernel<<<(int)(((long)N * 32 + TB - 1) / TB), TB, 0, stream>>>(
        sid, cid, pid, se, ce, pe, x0, N);
    // 2. per-relation in-degree (same for both layers), then invert in place
    count_kernel<<<(E + TB - 1) / TB, TB, 0, stream>>>(edst, etyp, cnt, E, N);
    inv_kernel<<<(int)(((long)N * 3 + TB - 1) / TB), TB, 0, stream>>>(cnt, (long)N * 3);

    const int gemmGrid   = (N + 31) / 32;
    const int scatGrid   = 3072;              // grid-stride; ~8 edges/lane + prefetch
    const long nh        = (long)N * 64;
    const int reluGrid   = (int)((nh + TB - 1) / TB);

    // ---- layer 1: K = 32 ----
    rgcn_gemm_kernel<<<gemmGrid, TB, 32 * 256 * sizeof(float), stream>>>(
        x0, root1, W1, b1, h1, z, N, 32);
    scatter_kernel<<<scatGrid, TB, 0, stream>>>(esrc, edst, etyp, z, cnt, h1, E, N);
    relu_kernel<<<reluGrid, TB, 0, stream>>>(h1, nh);

    // ---- layer 2: K = 64 ----
    rgcn_gemm_kernel<<<gemmGrid, TB, 64 * 256 * sizeof(float), stream>>>(
        h1, root2, W2, b2, h2, z, N, 64);
    scatter_kernel<<<scatGrid, TB, 0, stream>>>(esrc, edst, etyp, z, cnt, h2, E, N);
    relu_kernel<<<reluGrid, TB, 0, stream>>>(h2, nh);

    // ---- pool + linear ----
    pool_kernel<<<reluGrid, TB, 0, stream>>>(h2, batch, psum, pcnt, N);
    final_kernel<<<(G + TB - 1) / TB, TB, 0, stream>>>(psum, pcnt, lw, lb, out, G);
}